// NUSpectralConv1D_61529701482701
// MI455X (gfx1250) — compile-verified
//
#include <hip/hip_runtime.h>

typedef __attribute__((ext_vector_type(2))) float v2f;
typedef __attribute__((ext_vector_type(8))) float v8f;

#define B_    16
#define C_    64
#define N_    8192
#define M_    257      // 2*128 + 1 modes, k = -128..128
#define MPAD  272      // 17 tiles of 16
#define KHALF 128
#define NSEG  32       // N split into 32 segments per (b, mode-tile)

// D = A(16x4 f32) * B(4x16 f32) + C(16x16 f32), exact fp32 (RNE)
__device__ __forceinline__ v8f wmma4(v2f a, v2f b, v8f c) {
    return __builtin_amdgcn_wmma_f32_16x16x4_f32(
        /*neg_a=*/false, a, /*neg_b=*/false, b,
        /*c_mod=*/(short)0, c, /*reuse_a=*/false, /*reuse_b=*/false);
}

// ---------------------------------------------------------------------------
// Prep: trapezoid quadrature weights wz(B,N) + zero the F accumulators.
// ---------------------------------------------------------------------------
__global__ void prep_kernel(const float* __restrict__ z,
                            float* __restrict__ wz,
                            float* __restrict__ Fr,
                            float* __restrict__ Fi) {
    int idx = blockIdx.x * blockDim.x + threadIdx.x;
    const int totF = B_ * C_ * MPAD;
    if (idx < totF) { Fr[idx] = 0.0f; Fi[idx] = 0.0f; }
    if (idx < B_ * N_) {
        int b = idx / N_, n = idx % N_;
        const float* zb = z + (size_t)b * N_;
        float w;
        if (n == 0)            w = 0.5f * (zb[1] - zb[0]);
        else if (n == N_ - 1)  w = 0.5f * (zb[N_ - 1] - zb[N_ - 2]);
        else                   w = 0.5f * (zb[n + 1] - zb[n - 1]);
        wz[idx] = fmaxf(w, 0.0f);
    }
}

// ---------------------------------------------------------------------------
// Forward NUFFT: F[c,m] += sum_n x[c,n]*wz[n]*{cos,-sin}(z[n]*k_m)
// One wave per (b, mode-tile, n-segment). Lane owns mode col m = lane&15.
// A tile: rows c (lane&15), K = vgpr + 2*(lane>=16)  -> n offsets.
// Atomic-reduce D tiles into Fr/Fi.
// ---------------------------------------------------------------------------
__global__ __launch_bounds__(32)
void fwd_kernel(const float* __restrict__ x,
                const float* __restrict__ z,
                const float* __restrict__ wz,
                float* __restrict__ Fr,
                float* __restrict__ Fi) {
    const int b    = blockIdx.z;
    const int mt   = blockIdx.y;           // 0..16
    const int seg  = blockIdx.x;           // 0..NSEG-1
    const int lane = threadIdx.x;
    const int lo   = lane & 15;
    const int hi   = lane >> 4;            // 0 or 1
    const int m    = mt * 16 + lo;         // mode column owned by this lane
    const float kf = (float)(m - KHALF);

    const int nspan = N_ / NSEG;
    const int nbeg  = seg * nspan;

    const float* xb = x  + (size_t)b * C_ * N_;
    const float* zb = z  + (size_t)b * N_;
    const float* wb = wz + (size_t)b * N_;

    v8f accC[4] = {}; v8f accS[4] = {};

    for (int n0 = nbeg; n0 < nbeg + nspan; n0 += 4) {
        const int na = n0 + 2 * hi;        // this lane's two K slots: na, na+1
        float z0 = zb[na], z1 = zb[na + 1];
        float w0 = wb[na], w1 = wb[na + 1];
        v2f bc, bs;
        bc.x = __cosf(z0 * kf); bc.y = __cosf(z1 * kf);
        bs.x = __sinf(z0 * kf); bs.y = __sinf(z1 * kf);
#pragma unroll
        for (int t = 0; t < 4; ++t) {      // 4 channel tiles -> C = 64
            const float* xr = xb + (size_t)(t * 16 + lo) * N_;
            v2f a; a.x = xr[na] * w0; a.y = xr[na + 1] * w1;
            accC[t] = wmma4(a, bc, accC[t]);
            accS[t] = wmma4(a, bs, accS[t]);
        }
    }

    if (m < M_) {
#pragma unroll
        for (int t = 0; t < 4; ++t) {
#pragma unroll
            for (int v = 0; v < 8; ++v) {
                int c = t * 16 + v + 8 * hi;           // D row layout
                size_t o = ((size_t)b * C_ + c) * MPAD + m;
                atomicAdd(&Fr[o],  accC[t][v]);
                atomicAdd(&Fi[o], -accS[t][v]);        // F_imag = -sum
            }
        }
    }
}

// ---------------------------------------------------------------------------
// Depthwise complex weighting + mode quadrature vk, in place (Fr,Fi -> Gr,Gi).
// Pad region (m >= M_) stays zero from prep_kernel.
// ---------------------------------------------------------------------------
__global__ void weight_kernel(const float* __restrict__ wr,
                              const float* __restrict__ wi,
                              float* __restrict__ Fr,
                              float* __restrict__ Fi) {
    int idx = blockIdx.x * blockDim.x + threadIdx.x;
    if (idx >= B_ * C_ * MPAD) return;
    int m = idx % MPAD;
    if (m >= M_) return;
    int c = (idx / MPAD) % C_;
    float fr = Fr[idx], fi = Fi[idx];
    float r = wr[c * M_ + m], i = wi[c * M_ + m];
    float vk = (m == 0 || m == M_ - 1) ? 0.5f : 1.0f;
    Fr[idx] = (fr * r - fi * i) * vk;
    Fi[idx] = (fr * i + fi * r) * vk;
}

// ---------------------------------------------------------------------------
// Inverse NUFFT: y[c,n] = (sum_m Gr*cos - Gi*sin) / (2*pi) + bias[c]
// One wave per (b, 16-sample tile). Lane owns sample col n = lane&15 so z_n
// is fixed across the whole mode loop. Pad modes contribute zero (Gr/Gi pad).
// ---------------------------------------------------------------------------
__global__ __launch_bounds__(32)
void inv_kernel(const float* __restrict__ z,
                const float* __restrict__ Gr,
                const float* __restrict__ Gi,
                const float* __restrict__ bias,
                float* __restrict__ y) {
    const int b    = blockIdx.y;
    const int nt   = blockIdx.x;           // 0..511
    const int lane = threadIdx.x;
    const int lo   = lane & 15;
    const int hi   = lane >> 4;
    const int n    = nt * 16 + lo;
    const float zn = z[(size_t)b * N_ + n];

    const float* Grb = Gr + (size_t)b * C_ * MPAD;
    const float* Gib = Gi + (size_t)b * C_ * MPAD;

    v8f accC[4] = {}; v8f accS[4] = {};

    for (int m0 = 0; m0 < MPAD; m0 += 4) {
        const int ma = m0 + 2 * hi;        // this lane's two K slots (modes)
        float k0 = (float)(ma - KHALF), k1 = (float)(ma + 1 - KHALF);
        v2f bc, bs;
        bc.x = __cosf(zn * k0); bc.y = __cosf(zn * k1);
        bs.x = __sinf(zn * k0); bs.y = __sinf(zn * k1);
#pragma unroll
        for (int t = 0; t < 4; ++t) {
            const float* gr = Grb + (size_t)(t * 16 + lo) * MPAD + ma;
            const float* gi = Gib + (size_t)(t * 16 + lo) * MPAD + ma;
            v2f ar; ar.x = gr[0]; ar.y = gr[1];
            v2f ai; ai.x = gi[0]; ai.y = gi[1];
            accC[t] = wmma4(ar, bc, accC[t]);
            accS[t] = wmma4(ai, bs, accS[t]);
        }
    }

    const float inv2pi = 0.15915494309189535f;
#pragma unroll
    for (int t = 0; t < 4; ++t) {
#pragma unroll
        for (int v = 0; v < 8; ++v) {
            int c = t * 16 + v + 8 * hi;
            y[((size_t)b * C_ + c) * N_ + nt * 16 + lo] =
                (accC[t][v] - accS[t][v]) * inv2pi + bias[c];
        }
    }
}

// ---------------------------------------------------------------------------
extern "C" void kernel_launch(void* const* d_in, const int* in_sizes, int n_in,
                              void* d_out, int out_size, void* d_ws, size_t ws_size,
                              hipStream_t stream) {
    const float* x    = (const float*)d_in[0];   // (B,C,N)
    const float* z    = (const float*)d_in[1];   // (B,N)
    const float* wr   = (const float*)d_in[2];   // (C,M)
    const float* wi   = (const float*)d_in[3];   // (C,M)
    const float* bias = (const float*)d_in[4];   // (1,C,1)
    float* y = (float*)d_out;                    // (B,C,N)

    float* ws = (float*)d_ws;
    float* wz = ws;                              // B*N
    float* Fr = wz + (size_t)B_ * N_;            // B*C*MPAD
    float* Fi = Fr + (size_t)B_ * C_ * MPAD;     // B*C*MPAD

    const int prepN = B_ * C_ * MPAD;            // >= B*N, covers both jobs
    hipLaunchKernelGGL(prep_kernel, dim3((prepN + 255) / 256), dim3(256), 0, stream,
                       z, wz, Fr, Fi);
    hipLaunchKernelGGL(fwd_kernel, dim3(NSEG, MPAD / 16, B_), dim3(32), 0, stream,
                       x, z, wz, Fr, Fi);
    hipLaunchKernelGGL(weight_kernel, dim3((B_ * C_ * MPAD + 255) / 256), dim3(256), 0, stream,
                       wr, wi, Fr, Fi);
    hipLaunchKernelGGL(inv_kernel, dim3(N_ / 16, B_), dim3(32), 0, stream,
                       z, Fr, Fi, bias, y);
}